// NGMConvLayer_86681029967973
// MI455X (gfx1250) — compile-verified
//
#include <hip/hip_runtime.h>
#include <hip/hip_bf16.h>

typedef __attribute__((ext_vector_type(2))) float v2f;
typedef __attribute__((ext_vector_type(4))) float v4f;
typedef __attribute__((ext_vector_type(8))) float v8f;

constexpr int NN = 2048;   // nodes
constexpr int EF = 16;     // edge features == out features
constexpr int INF = 64;    // in node features
constexpr float EPSV = 1e-8f;

#if defined(__has_builtin)
#if __has_builtin(__builtin_amdgcn_wmma_f32_16x16x4_f32)
#define HAVE_WMMA_F32X4 1
#endif
#endif

// ---------------------------------------------------------------------------
// Kernel 1: both MLPs via V_WMMA_F32_16X16X4_F32.
// One wave32 computes a 16-row output tile (16x16).
//   A-frag (16x4 f32): lane l -> row = l&15 ; VGPR v -> K = v + 2*(l>>4)
//   B-frag (4x16 f32): lane l -> col = l&15 ; VGPR v -> K = v + 2*(l>>4)
//   C/D   (16x16 f32): lane l -> N = l&15 ; VGPR r -> M = r + 8*(l>>4)
// Layer 1: X(16x64) @ w1(64x16) + b1, relu  (16 WMMA, K-chunks of 4)
// Layer 2: H(16x16) @ w2(16x16) + b2, relu  (4 WMMA, H re-staged via LDS)
// blockDim = 128 (4 waves), grid = 2048/16/4 = 32 blocks.
// ---------------------------------------------------------------------------
__global__ __launch_bounds__(128) void ngm_mlp_kernel(
    const float* __restrict__ x,
    const float* __restrict__ w1, const float* __restrict__ b1,
    const float* __restrict__ w2, const float* __restrict__ b2,
    const float* __restrict__ sw1, const float* __restrict__ sb1,
    const float* __restrict__ sw2, const float* __restrict__ sb2,
    float* __restrict__ x1out, float* __restrict__ selfout)
{
#ifdef HAVE_WMMA_F32X4
  __shared__ float hlds[4 * 256];          // one 16x16 f32 tile per wave

  const int wave = threadIdx.x >> 5;
  const int lane = threadIdx.x & 31;
  const int tile = blockIdx.x * 4 + wave;  // 16-row tile index
  const int ln   = lane & 15;              // row (for A) / col (for B,C,D)
  const int grp  = lane >> 4;
  const int koff = grp << 1;
  const int grow = tile * 16 + ln;         // global row for A-frags
  float* hl = hlds + wave * 256;

  for (int p = 0; p < 2; ++p) {
    const float* cw1 = p ? sw1 : w1;
    const float* cb1 = p ? sb1 : b1;
    const float* cw2 = p ? sw2 : w2;
    const float* cb2 = p ? sb2 : b2;
    float* dst       = p ? selfout : x1out;

    // ---- layer 1: C = X(16x64) @ cw1(64x16) + cb1 ----
    v8f c;
    {
      float bias = cb1[ln];
#pragma unroll
      for (int r = 0; r < 8; ++r) c[r] = bias;
    }
#pragma unroll
    for (int kc = 0; kc < 16; ++kc) {
      const int kb = kc * 4 + koff;
      v2f a, b;
      a.x = x[grow * INF + kb];
      a.y = x[grow * INF + kb + 1];
      b.x = cw1[(kb    ) * EF + ln];
      b.y = cw1[(kb + 1) * EF + ln];
      c = __builtin_amdgcn_wmma_f32_16x16x4_f32(false, a, false, b,
                                                (short)0, c, false, false);
    }
#pragma unroll
    for (int r = 0; r < 8; ++r) c[r] = fmaxf(c[r], 0.0f);

    // stage H into LDS so we can regather it in A-frag layout
#pragma unroll
    for (int r = 0; r < 8; ++r) hl[(r + 8 * grp) * EF + ln] = c[r];
    __syncthreads();

    // ---- layer 2: C2 = H(16x16) @ cw2(16x16) + cb2 ----
    v8f c2;
    {
      float bias2 = cb2[ln];
#pragma unroll
      for (int r = 0; r < 8; ++r) c2[r] = bias2;
    }
#pragma unroll
    for (int kc = 0; kc < 4; ++kc) {
      const int kb = kc * 4 + koff;
      v2f a, b;
      a.x = hl[ln * EF + kb];
      a.y = hl[ln * EF + kb + 1];
      b.x = cw2[(kb    ) * EF + ln];
      b.y = cw2[(kb + 1) * EF + ln];
      c2 = __builtin_amdgcn_wmma_f32_16x16x4_f32(false, a, false, b,
                                                 (short)0, c2, false, false);
    }
#pragma unroll
    for (int r = 0; r < 8; ++r) {
      float v = fmaxf(c2[r], 0.0f);
      dst[(tile * 16 + r + 8 * grp) * EF + ln] = v;
    }
    __syncthreads();   // before LDS reuse by next pass
  }
#else
  // Fallback (no f32x4 WMMA builtin): plain VALU MLP, grid-stride.
  int gid    = blockIdx.x * blockDim.x + threadIdx.x;
  int stride = gridDim.x * blockDim.x;
  for (int idx = gid; idx < NN * EF; idx += stride) {
    int row = idx / EF, col = idx % EF;
    for (int p = 0; p < 2; ++p) {
      const float* cw1 = p ? sw1 : w1;
      const float* cb1 = p ? sb1 : b1;
      const float* cw2 = p ? sw2 : w2;
      const float* cb2 = p ? sb2 : b2;
      float* dst       = p ? selfout : x1out;
      float acc = cb2[col];
      for (int h = 0; h < EF; ++h) {
        float hv = cb1[h];
        for (int k = 0; k < INF; ++k) hv += x[row * INF + k] * cw1[k * EF + h];
        hv = fmaxf(hv, 0.0f);
        acc += hv * cw2[h * EF + col];
      }
      dst[idx] = fmaxf(acc, 0.0f);
    }
  }
#endif
}

// ---------------------------------------------------------------------------
// Kernel 2: fused   outW = W (non-temporal stream copy)
//                   x2[i,e] = (1/(rowsum(A,i)+eps)) * sum_j A[i,j]*W[i,j,e]*x1[j,e]
//                           + selfMLP[i,e]
// One block per row i; 256 threads; thread t covers e-chunk (t&3, 4 floats)
// and j = (t>>2) + 64*it.  Rowsum of A computed in-flight (each j seen by the
// 4 e-chunk lanes -> sum of partials == 4 * rowsum).
// ---------------------------------------------------------------------------
__global__ __launch_bounds__(256) void ngm_einsum_copy_kernel(
    const float* __restrict__ A, const float* __restrict__ W,
    const float* __restrict__ x1ws, const float* __restrict__ selfws,
    float* __restrict__ outW, float* __restrict__ outx2)
{
  const int i   = blockIdx.x;
  const int tid = threadIdx.x;
  const int e4  = tid & 3;    // which float4 chunk of E
  const int js  = tid >> 2;   // 0..63 j slot

  const size_t rowW  = (size_t)i * (NN * EF);
  const float* Arow  = A + (size_t)i * NN;

  v4f acc = {0.f, 0.f, 0.f, 0.f};
  float asum = 0.f;

#pragma unroll 8
  for (int it = 0; it < NN / 64; ++it) {
    const int j = js + (it << 6);
    const size_t idx = rowW + (size_t)(j * EF + (e4 << 2));
    v4f w = __builtin_nontemporal_load((const v4f*)(W + idx));
    __builtin_nontemporal_store(w, (v4f*)(outW + idx));
    const float a = Arow[j];
    const v4f xv = *(const v4f*)(x1ws + j * EF + (e4 << 2));
    acc  += (a * xv) * w;
    asum += a;
  }

  __shared__ v4f   red[256];
  __shared__ float redA[256];
  red[tid]  = acc;
  redA[tid] = asum;
  __syncthreads();

  // tree reduce over j-slots; stride multiples of 4 preserve the e-chunk id
  for (int off = 128; off >= 4; off >>= 1) {
    if (tid < off) {
      red[tid]  += red[tid + off];
      redA[tid] += redA[tid + off];
    }
    __syncthreads();
  }
  if (tid == 0) redA[0] = redA[0] + redA[1] + redA[2] + redA[3];
  __syncthreads();

  if (tid < EF) {
    const float rowsum = redA[0] * 0.25f;     // each j counted by 4 lanes
    const float val    = red[tid >> 2][tid & 3];
    outx2[i * EF + tid] = val / (rowsum + EPSV) + selfws[i * EF + tid];
  }
}

extern "C" void kernel_launch(void* const* d_in, const int* in_sizes, int n_in,
                              void* d_out, int out_size, void* d_ws, size_t ws_size,
                              hipStream_t stream) {
  const float* A   = (const float*)d_in[0];
  const float* W   = (const float*)d_in[1];
  const float* x   = (const float*)d_in[2];
  const float* w1  = (const float*)d_in[3];
  const float* b1  = (const float*)d_in[4];
  const float* w2  = (const float*)d_in[5];
  const float* b2  = (const float*)d_in[6];
  const float* sw1 = (const float*)d_in[7];
  const float* sb1 = (const float*)d_in[8];
  const float* sw2 = (const float*)d_in[9];
  const float* sb2 = (const float*)d_in[10];

  float* outW  = (float*)d_out;                       // (N,N,E) pass-through
  float* outx2 = outW + (size_t)NN * NN * EF;         // (N,E)

  float* x1ws   = (float*)d_ws;                       // (N,E)
  float* selfws = x1ws + NN * EF;                     // (N,E)

  // MLPs (WMMA): 128 tiles of 16 rows, 4 waves/block -> 32 blocks x 128 thr
  ngm_mlp_kernel<<<NN / 16 / 4, 128, 0, stream>>>(
      x, w1, b1, w2, b2, sw1, sb1, sw2, sb2, x1ws, selfws);

  // Fused einsum + W copy + normalize + self-add: one block per row
  ngm_einsum_copy_kernel<<<NN, 256, 0, stream>>>(
      A, W, x1ws, selfws, outW, outx2);
}